// EmformerLayer_22840636080587
// MI455X (gfx1250) — compile-verified
//
#include <hip/hip_runtime.h>

typedef __attribute__((ext_vector_type(2))) float v2f;
typedef __attribute__((ext_vector_type(8))) float v8f;

#define BD 8       // batch
#define DD 512     // model dim
#define dH 64      // head dim
#define RR 32
#define UU 1024
#define MM 32
#define QQ 1088    // R+U+S
#define KVK 1088   // M+R+U
#define TRU 1056   // R+U
#define SROW 1092  // padded LDS row stride (1088+4) -> conflict-free column walks

static __device__ __forceinline__ v8f wmma4(v2f a, v2f b, v8f c) {
  // D = A(16x4 f32) * B(4x16 f32) + C  -- full f32 matrix pipe
  return __builtin_amdgcn_wmma_f32_16x16x4_f32(false, a, false, b, (short)0, c,
                                               false, false);
}

// ---------------------------------------------------------------------------
// Generic 16x16-tile f32 WMMA GEMM:  C[r,o] = alpha*(sum_c A[r,c]*W[o,c] + bias[o]) (+res)
// Row r = t*8+b; row source gathered from up to 3 segment base pointers by t.
// K = DD = 512, lda = ldw = ldc = 512. One wave per tile.
// ---------------------------------------------------------------------------
__global__ __launch_bounds__(32) void gemm16_f32(
    const float* __restrict__ A0, const float* __restrict__ A1,
    const float* __restrict__ A2, int seg0, int seg1,
    const float* __restrict__ W, const float* __restrict__ bias,
    const float* __restrict__ residual, float alpha, float* __restrict__ C) {
  const int lane = threadIdx.x;
  const int m  = lane & 15;          // M (A) / N (B) index for this lane
  const int kb = (lane >> 4) << 1;   // K sub-offset {0,2}
  const int row = blockIdx.x * 16 + m;
  const int t = row >> 3;
  const int b = row & 7;
  const float* arow;
  if (t < seg0)      arow = A0 + ((size_t)t * BD + b) * DD;
  else if (t < seg1) arow = A1 + ((size_t)(t - seg0) * BD + b) * DD;
  else               arow = A2 + ((size_t)(t - seg1) * BD + b) * DD;
  const int o0 = blockIdx.y * 16;
  const float* wrow = W + (size_t)(o0 + m) * DD;

  v8f acc = {0.f, 0.f, 0.f, 0.f, 0.f, 0.f, 0.f, 0.f};
#pragma unroll 8
  for (int c0 = 0; c0 < DD; c0 += 4) {
    v2f a  = *(const v2f*)(arow + c0 + kb);
    v2f bb = *(const v2f*)(wrow + c0 + kb);
    acc = wmma4(a, bb, acc);
  }

  const int col = o0 + m;
  const int rbase = blockIdx.x * 16 + ((lane >> 4) << 3);
  const float bv = bias ? bias[col] : 0.f;
#pragma unroll
  for (int i = 0; i < 8; ++i) {
    const int r = rbase + i;
    float v = alpha * (acc[i] + bv);
    if (residual) v += residual[(size_t)r * DD + col];
    C[(size_t)r * DD + col] = v;
  }
}

// ---------------------------------------------------------------------------
// pw1 + GLU fused: each wave computes paired channel tiles (o, o+512) of
// y = X @ w_pw1^T + b_pw1 and emits h1 = y_a * sigmoid(y_g).
// ---------------------------------------------------------------------------
__global__ __launch_bounds__(32) void pw1_glu(
    const float* __restrict__ X, const float* __restrict__ Wp,
    const float* __restrict__ bp, float* __restrict__ H1) {
  const int lane = threadIdx.x;
  const int m  = lane & 15;
  const int kb = (lane >> 4) << 1;
  const int row = blockIdx.x * 16 + m;
  const float* arow = X + (size_t)row * DD;
  const int o0 = blockIdx.y * 16;
  const float* wa = Wp + (size_t)(o0 + m) * DD;
  const float* wg = Wp + (size_t)(o0 + m + DD) * DD;

  v8f accA = {0.f, 0.f, 0.f, 0.f, 0.f, 0.f, 0.f, 0.f};
  v8f accG = {0.f, 0.f, 0.f, 0.f, 0.f, 0.f, 0.f, 0.f};
#pragma unroll 4
  for (int c0 = 0; c0 < DD; c0 += 4) {
    v2f a = *(const v2f*)(arow + c0 + kb);
    accA = wmma4(a, *(const v2f*)(wa + c0 + kb), accA);
    accG = wmma4(a, *(const v2f*)(wg + c0 + kb), accG);
  }

  const int col = o0 + m;
  const int rbase = blockIdx.x * 16 + ((lane >> 4) << 3);
  const float ba = bp[col], bg = bp[col + DD];
#pragma unroll
  for (int i = 0; i < 8; ++i) {
    float va = accA[i] + ba;
    float vg = accG[i] + bg;
    float h = va * (1.f / (1.f + __expf(-vg)));
    H1[(size_t)(rbase + i) * DD + col] = h;
  }
}

// ---------------------------------------------------------------------------
// Attention: one wave per (head bh, 16-row q tile). Full 16x1088 score strip
// in LDS (~70KB, fits 320KB WGP LDS); analytic Emformer chunk mask + length
// padding; two-phase softmax; p@V via f32 WMMA reading p from LDS.
// ---------------------------------------------------------------------------
__global__ __launch_bounds__(32) void attn16(
    const float* __restrict__ Qb, const float* __restrict__ Kb,
    const float* __restrict__ Vb, const int* __restrict__ lengths,
    float* __restrict__ Ab) {
  extern __shared__ float smem[];
  float* sc   = smem;                 // [16][SROW]
  float* red1 = smem + 16 * SROW;     // [32]
  float* red2 = red1 + 32;            // [32]

  const int lane = threadIdx.x;
  const int bh = blockIdx.y;
  const int b = bh >> 3, h = bh & 7;
  const int q0 = blockIdx.x * 16;
  const int m  = lane & 15;
  const int hi = lane >> 4;
  const int kb = hi << 1;
  const int limit = MM + RR + lengths[b];  // cols >= limit are padded out

  const float* qrow = Qb + ((size_t)(q0 + m) * BD + b) * DD + h * dH;

  // ---- scores: 68 KV tiles of 16, K=64 contraction ----
  for (int kt = 0; kt < KVK / 16; ++kt) {
    const int kv0 = kt * 16;
    const float* krow = Kb + ((size_t)(kv0 + m) * BD + b) * DD + h * dH;
    v8f acc = {0.f, 0.f, 0.f, 0.f, 0.f, 0.f, 0.f, 0.f};
#pragma unroll
    for (int c0 = 0; c0 < dH; c0 += 4) {
      v2f a  = *(const v2f*)(qrow + c0 + kb);
      v2f bb = *(const v2f*)(krow + c0 + kb);
      acc = wmma4(a, bb, acc);
    }
    const int col = kv0 + m;
#pragma unroll
    for (int i = 0; i < 8; ++i) {
      const int qq = q0 + i + (hi << 3);               // q row (< TRU)
      const int c = (qq < RR) ? qq : ((qq - RR) >> 5); // chunk id
      bool ok;
      if (col < MM) {
        ok = col < c;                                  // memory banks < c
      } else if (col < MM + RR) {
        ok = (col - MM) == c;                          // own right-context col
      } else {
        const int u = col - MM - RR;
        const int lo = (c > 0 ? c - 1 : 0) << 5;
        const int hi2 = (c + 1) << 5;
        ok = (u >= lo) && (u < hi2);                   // own + 1 left chunk
      }
      ok = ok && (col < limit);
      sc[(size_t)(i + (hi << 3)) * SROW + col] = ok ? acc[i] : -1.0e8f;
    }
  }
  __syncthreads();

  // ---- softmax: lane handles row m, half hi (544 cols each) ----
  {
    const int base = m * SROW + hi * (KVK / 2);
    float mx = -3.0e38f;
    for (int j = 0; j < KVK / 2; ++j) mx = fmaxf(mx, sc[base + j]);
    red1[lane] = mx;
    __syncthreads();
    mx = fmaxf(red1[m], red1[m + 16]);
    float sum = 0.f;
    for (int j = 0; j < KVK / 2; ++j) {
      float e = __expf(sc[base + j] - mx);
      sc[base + j] = e;
      sum += e;
    }
    red2[lane] = sum;
    __syncthreads();
    const float inv = 1.f / (red2[m] + red2[m + 16]);
    for (int j = 0; j < KVK / 2; ++j) sc[base + j] *= inv;
  }
  __syncthreads();

  // ---- attn = p @ V : 4 n-tiles of 16, K=1088 contraction ----
  const float* vbase = Vb + (size_t)b * DD + h * dH;
  for (int nt = 0; nt < 4; ++nt) {
    const int n0 = nt * 16;
    const float* vcol = vbase + n0 + m;
    v8f acc = {0.f, 0.f, 0.f, 0.f, 0.f, 0.f, 0.f, 0.f};
#pragma unroll 4
    for (int k0 = 0; k0 < KVK; k0 += 4) {
      v2f a  = { sc[m * SROW + k0 + kb], sc[m * SROW + k0 + kb + 1] };
      v2f bb = { vcol[(size_t)(k0 + kb) * (BD * DD)],
                 vcol[(size_t)(k0 + kb + 1) * (BD * DD)] };
      acc = wmma4(a, bb, acc);
    }
    const int col = h * dH + n0 + m;
#pragma unroll
    for (int i = 0; i < 8; ++i) {
      const int qq = q0 + i + (hi << 3);
      Ab[((size_t)qq * BD + b) * DD + col] = acc[i];
    }
  }
}

// ---------------------------------------------------------------------------
// Depthwise 31-tap conv along t (per b,c), zero padded, + bias.
// ---------------------------------------------------------------------------
__global__ __launch_bounds__(256) void dwconv31(
    const float* __restrict__ H1v, const float* __restrict__ wdw,
    const float* __restrict__ bdw, float* __restrict__ H2v) {
  const int idx = blockIdx.x * 256 + threadIdx.x;  // (t*8+b)*512 + c
  const int c = idx & 511;
  const int tb = idx >> 9;
  const int b = tb & 7;
  const int t = tb >> 3;
  float acc = bdw[c];
#pragma unroll
  for (int j = 0; j < 31; ++j) {
    const int tt = t + j - 15;
    if (tt >= 0 && tt < TRU)
      acc += H1v[((size_t)tt * BD + b) * DD + c] * wdw[c * 31 + j];
  }
  H2v[idx] = acc;
}

// ---------------------------------------------------------------------------
// LayerNorm over channel dim (512) per (t,b) row, then SiLU.
// ---------------------------------------------------------------------------
__global__ __launch_bounds__(256) void ln_silu(
    const float* __restrict__ X, const float* __restrict__ g,
    const float* __restrict__ be, float* __restrict__ Y) {
  __shared__ float red[256];
  const int row = blockIdx.x;
  const int tid = threadIdx.x;
  const float* x = X + (size_t)row * DD;
  const float v0 = x[tid], v1 = x[tid + 256];

  red[tid] = v0 + v1;
  __syncthreads();
  for (int off = 128; off; off >>= 1) {
    if (tid < off) red[tid] += red[tid + off];
    __syncthreads();
  }
  const float mu = red[0] * (1.f / 512.f);
  __syncthreads();

  const float d0 = v0 - mu, d1 = v1 - mu;
  red[tid] = d0 * d0 + d1 * d1;
  __syncthreads();
  for (int off = 128; off; off >>= 1) {
    if (tid < off) red[tid] += red[tid + off];
    __syncthreads();
  }
  const float rs = rsqrtf(red[0] * (1.f / 512.f) + 1e-5f);

  const float y0 = d0 * rs * g[tid] + be[tid];
  const float y1 = d1 * rs * g[tid + 256] + be[tid + 256];
  Y[(size_t)row * DD + tid]       = y0 * (1.f / (1.f + __expf(-y0)));
  Y[(size_t)row * DD + tid + 256] = y1 * (1.f / (1.f + __expf(-y1)));
}

// ---------------------------------------------------------------------------
extern "C" void kernel_launch(void* const* d_in, const int* in_sizes, int n_in,
                              void* d_out, int out_size, void* d_ws,
                              size_t ws_size, hipStream_t stream) {
  (void)in_sizes; (void)n_in; (void)out_size; (void)ws_size;
  const float* utter = (const float*)d_in[0];
  const float* rc    = (const float*)d_in[1];
  const float* summ  = (const float*)d_in[2];
  const float* mem   = (const float*)d_in[3];
  const float* w_q   = (const float*)d_in[4];
  const float* b_q   = (const float*)d_in[5];
  const float* w_kv  = (const float*)d_in[6];
  const float* b_kv  = (const float*)d_in[7];
  const float* w_out = (const float*)d_in[8];
  const float* b_out = (const float*)d_in[9];
  const float* w_pw1 = (const float*)d_in[10];
  const float* b_pw1 = (const float*)d_in[11];
  const float* w_dw  = (const float*)d_in[12];
  const float* b_dw  = (const float*)d_in[13];
  const float* ln_g  = (const float*)d_in[14];
  const float* ln_b  = (const float*)d_in[15];
  const float* w_pw2 = (const float*)d_in[16];
  const float* b_pw2 = (const float*)d_in[17];
  const int*   lens  = (const int*)d_in[18];
  // d_in[19] (attention_mask) recomputed analytically on device.

  float* ws = (float*)d_ws;
  const size_t BUF = (size_t)QQ * BD * DD;  // 4,456,448 floats per region
  float* qbuf  = ws;            // -> reused as out_ru
  float* kbuf  = ws + BUF;      // -> reused as h1
  float* vbuf  = ws + 2 * BUF;  // -> reused as h2
  float* abuf  = ws + 3 * BUF;  // -> reused as h3
  float* outru = qbuf;
  float* h1 = kbuf;
  float* h2 = vbuf;
  float* h3 = abuf;
  float* out = (float*)d_out;

  const dim3 wave(32);
  const int BIG = 1 << 20;

  // Q projection (scaled by d^-0.5 = 0.125): rows = 1088*8 -> 544 tiles
  gemm16_f32<<<dim3(544, 32), wave, 0, stream>>>(
      rc, utter, summ, RR, RR + UU, w_q, b_q, nullptr, 0.125f, qbuf);
  // K / V projections (split halves of w_kv / b_kv)
  gemm16_f32<<<dim3(544, 32), wave, 0, stream>>>(
      mem, rc, utter, MM, MM + RR, w_kv, b_kv, nullptr, 1.0f, kbuf);
  gemm16_f32<<<dim3(544, 32), wave, 0, stream>>>(
      mem, rc, utter, MM, MM + RR, w_kv + (size_t)DD * DD, b_kv + DD, nullptr,
      1.0f, vbuf);

  // Attention: 66 q-tiles (only rows < R+U needed) x 64 heads
  const size_t lds_bytes = (size_t)(16 * SROW + 64) * sizeof(float);
  attn16<<<dim3(66, 64), wave, lds_bytes, stream>>>(qbuf, kbuf, vbuf, lens,
                                                    abuf);

  // Output projection -> out_ru (reuses query region)
  gemm16_f32<<<dim3(528, 32), wave, 0, stream>>>(
      abuf, abuf, abuf, BIG, 2 * BIG, w_out, b_out, nullptr, 1.0f, outru);

  // Conv module
  pw1_glu<<<dim3(528, 32), wave, 0, stream>>>(outru, w_pw1, b_pw1, h1);
  dwconv31<<<dim3(16896), dim3(256), 0, stream>>>(h1, w_dw, b_dw, h2);
  ln_silu<<<dim3(8448), dim3(256), 0, stream>>>(h2, ln_g, ln_b, h3);

  // pw2 + residual(out_ru) -> d_out
  gemm16_f32<<<dim3(528, 32), wave, 0, stream>>>(
      h3, h3, h3, BIG, 2 * BIG, w_pw2, b_pw2, outru, 1.0f, out);
}